// DualNetworkGNN_80590766342428
// MI455X (gfx1250) — compile-verified
//
#include <hip/hip_runtime.h>
#include <hip/hip_bf16.h>

// CDNA5 / gfx1250. wave32. fp32 WMMA (V_WMMA_F32_16X16X4_F32) for all GEMMs,
// global_atomic_add_f32 for all graph segment-sums. Everything L2-resident.

typedef __attribute__((ext_vector_type(2))) float v2f;
typedef __attribute__((ext_vector_type(8))) float v8f;

// ---------------------------------------------------------------------------
// Generic fused GEMM:  C[M x 128] = act( A1[M x K1] @ W1[K1 x 128] * scale1
//                                      + A2[M x K2] @ W2[K2 x 128] + bias )
// scale1 = cscale1 * (rowscale1 ? rowscale1[row] : 1)   (per-row, pair 1 only)
// Block = 128 threads = 4 waves; wave w owns rows [ (blockIdx*4+w)*16 , +16 ).
//
// W is staged in LDS in a k-interleaved layout:
//    element (k, col)  ->  lw[(k>>2)*512 + col*4 + (k&3)]
// so the fp32-WMMA B fragment {W[kk][col], W[kk+1][col]} (kk = k + 2*lhalf)
// is one contiguous 8-byte ds_load_b64 straight into an even VGPR pair
// (compiler merges pairs into ds_load_2addr_b64), with per-tile immediate
// offsets and a conflict-free bank pattern (lane halves hit disjoint banks).
//
// The scale path is split into separate k-loops (wave-uniform, loop-invariant
// condition) so unscaled GEMMs get a pure ds_load+wmma inner loop with no
// per-step v_pk_mul / v_cndmask.
//
// fp32 WMMA fragment layout (ISA 7.12.2):
//   A (16x4):  lanes 0-15 hold M=lane, {K=k,k+1}; lanes 16-31 hold {K=k+2,k+3}
//   B (4x16):  VGPR0 = W[k + 2*(lane/16)][N=lane%16], VGPR1 = next K row
//   C (16x16): acc[j] -> M = j + 8*(lane/16), N = lane%16
// ---------------------------------------------------------------------------
__global__ void __launch_bounds__(128)
gemm128_wmma(const float* __restrict__ A1, int K1, const float* __restrict__ W1,
             const float* __restrict__ A2, int K2, const float* __restrict__ W2,
             const float* __restrict__ bias, const float* __restrict__ rowscale1,
             float cscale1, int relu, float* __restrict__ C, int M)
{
    extern __shared__ float lw[];
    const int tid   = threadIdx.x;
    const int lane  = tid & 31;
    const int wave  = tid >> 5;
    const int lrow  = lane & 15;
    const int lhalf = lane >> 4;
    const int row0  = (blockIdx.x * 4 + wave) * 16;
    const bool active = row0 < M;          // wave-uniform -> EXEC all ones inside
    const int row   = row0 + lrow;

    v8f acc[8] = {};

    // ---- pair 1 ----
    {
        const int col = tid;               // blockDim == 128
        for (int g = 0; g < (K1 >> 2); ++g) {
            const float* wsrc = W1 + (size_t)(g << 2) * 128 + col;
            float4 v;
            v.x = wsrc[0]; v.y = wsrc[128]; v.z = wsrc[256]; v.w = wsrc[384];
            *(float4*)(lw + g * 512 + col * 4) = v;
        }
        __syncthreads();
        if (active) {
            const bool doscale = (rowscale1 != nullptr) || (cscale1 != 1.0f);
            const float* arow = A1 + (size_t)row * K1;
            if (doscale) {
                const float s = cscale1 * (rowscale1 ? rowscale1[row] : 1.0f);
                for (int k = 0; k < K1; k += 4) {
                    v2f a = *(const v2f*)(arow + k + 2 * lhalf);
                    a.x *= s; a.y *= s;
                    const float* bk = lw + ((k >> 2) * 512 + lrow * 4 + 2 * lhalf);
#pragma unroll
                    for (int t = 0; t < 8; ++t) {
                        const v2f b = *(const v2f*)(bk + t * 64);
                        acc[t] = __builtin_amdgcn_wmma_f32_16x16x4_f32(
                            false, a, false, b, (short)0, acc[t], false, false);
                    }
                }
            } else {
                for (int k = 0; k < K1; k += 4) {
                    const v2f a = *(const v2f*)(arow + k + 2 * lhalf);
                    const float* bk = lw + ((k >> 2) * 512 + lrow * 4 + 2 * lhalf);
#pragma unroll
                    for (int t = 0; t < 8; ++t) {
                        const v2f b = *(const v2f*)(bk + t * 64);
                        acc[t] = __builtin_amdgcn_wmma_f32_16x16x4_f32(
                            false, a, false, b, (short)0, acc[t], false, false);
                    }
                }
            }
        }
    }
    // ---- pair 2 (optional) ----
    if (K2 > 0) {
        __syncthreads();
        const int col = tid;
        for (int g = 0; g < (K2 >> 2); ++g) {
            const float* wsrc = W2 + (size_t)(g << 2) * 128 + col;
            float4 v;
            v.x = wsrc[0]; v.y = wsrc[128]; v.z = wsrc[256]; v.w = wsrc[384];
            *(float4*)(lw + g * 512 + col * 4) = v;
        }
        __syncthreads();
        if (active) {
            const float* arow = A2 + (size_t)row * K2;
            for (int k = 0; k < K2; k += 4) {
                const v2f a = *(const v2f*)(arow + k + 2 * lhalf);
                const float* bk = lw + ((k >> 2) * 512 + lrow * 4 + 2 * lhalf);
#pragma unroll
                for (int t = 0; t < 8; ++t) {
                    const v2f b = *(const v2f*)(bk + t * 64);
                    acc[t] = __builtin_amdgcn_wmma_f32_16x16x4_f32(
                        false, a, false, b, (short)0, acc[t], false, false);
                }
            }
        }
    }
    // ---- epilogue ----
    if (active) {
#pragma unroll
        for (int t = 0; t < 8; ++t) {
            const int col = t * 16 + lrow;
            const float bv = bias ? bias[col] : 0.0f;
#pragma unroll
            for (int j = 0; j < 8; ++j) {
                float v = acc[t][j] + bv;
                if (relu) v = fmaxf(v, 0.0f);
                C[(size_t)(row0 + j + 8 * lhalf) * 128 + col] = v;
            }
        }
    }
}

// ---------------------------------------------------------------------------
// Graph kernels (fp32 HW atomics -> global_atomic_add_f32)
// ---------------------------------------------------------------------------
__global__ void deg_ub_kernel(const int* __restrict__ src, const int* __restrict__ dst,
                              float* __restrict__ deg, int E, int U)
{
    int e = blockIdx.x * blockDim.x + threadIdx.x;
    if (e < E) {
        unsafeAtomicAdd(deg + src[e], 1.0f);
        unsafeAtomicAdd(deg + U + dst[e], 1.0f);
    }
}

__global__ void count_kernel(const int* __restrict__ idx, float* __restrict__ cnt, int E)
{
    int e = blockIdx.x * blockDim.x + threadIdx.x;
    if (e < E) unsafeAtomicAdd(cnt + idx[e], 1.0f);
}

__global__ void dinv_kernel(const float* __restrict__ deg, float* __restrict__ dinv, int n)
{
    int i = blockIdx.x * blockDim.x + threadIdx.x;
    if (i < n) { float d = deg[i]; dinv[i] = d > 0.0f ? rsqrtf(d) : 0.0f; }
}

__global__ void invmax1_kernel(const float* __restrict__ d, float* __restrict__ o, int n)
{
    int i = blockIdx.x * blockDim.x + threadIdx.x;
    if (i < n) o[i] = 1.0f / fmaxf(d[i], 1.0f);
}

__global__ void vec_add_kernel(float* __restrict__ a, const float* __restrict__ b, size_t n4)
{
    size_t i = (size_t)blockIdx.x * blockDim.x + threadIdx.x;
    if (i < n4) {
        float4 x = ((const float4*)b)[i];
        float4 y = ((float4*)a)[i];
        y.x += x.x; y.y += x.y; y.z += x.z; y.w += x.w;
        ((float4*)a)[i] = y;
    }
}

// One wave per undirected user-book edge; both directions with symmetric norm.
__global__ void prop_ub_kernel(const float* __restrict__ cur, float* __restrict__ nxt,
                               const int* __restrict__ src, const int* __restrict__ dst,
                               const float* __restrict__ dinv, int E, int U)
{
    const int lane = threadIdx.x & 31;
    const int e = (int)(((size_t)blockIdx.x * blockDim.x + threadIdx.x) >> 5);
    if (e >= E) return;
    const int u = src[e];
    const int b = U + dst[e];
    const float nrm = dinv[u] * dinv[b];
    const float4 xu = *(const float4*)(cur + (size_t)u * 128 + lane * 4);
    const float4 xb = *(const float4*)(cur + (size_t)b * 128 + lane * 4);
    float* pb = nxt + (size_t)b * 128 + lane * 4;
    float* pu = nxt + (size_t)u * 128 + lane * 4;
    unsafeAtomicAdd(pb + 0, xu.x * nrm); unsafeAtomicAdd(pb + 1, xu.y * nrm);
    unsafeAtomicAdd(pb + 2, xu.z * nrm); unsafeAtomicAdd(pb + 3, xu.w * nrm);
    unsafeAtomicAdd(pu + 0, xb.x * nrm); unsafeAtomicAdd(pu + 1, xb.y * nrm);
    unsafeAtomicAdd(pu + 2, xb.z * nrm); unsafeAtomicAdd(pu + 3, xb.w * nrm);
}

// One wave per book-book edge: agg[dst] += h[src]
__global__ void scat_bb_kernel(const float* __restrict__ h, float* __restrict__ agg,
                               const int* __restrict__ src, const int* __restrict__ dst, int E)
{
    const int lane = threadIdx.x & 31;
    const int e = (int)(((size_t)blockIdx.x * blockDim.x + threadIdx.x) >> 5);
    if (e >= E) return;
    const int s = src[e], d = dst[e];
    const float4 x = *(const float4*)(h + (size_t)s * 128 + lane * 4);
    float* p = agg + (size_t)d * 128 + lane * 4;
    unsafeAtomicAdd(p + 0, x.x); unsafeAtomicAdd(p + 1, x.y);
    unsafeAtomicAdd(p + 2, x.z); unsafeAtomicAdd(p + 3, x.w);
}

// One wave per book: 2-way attention softmax + fusion.
__global__ void attn_fuse_kernel(const float* __restrict__ T, const float* __restrict__ Wf2,
                                 const float* __restrict__ bf2, const float* __restrict__ collab,
                                 const float* __restrict__ content, float* __restrict__ fused, int B)
{
    const int lane = threadIdx.x & 31;
    const int r = (int)(((size_t)blockIdx.x * blockDim.x + threadIdx.x) >> 5);
    if (r >= B) return;
    const float* trow = T + (size_t)r * 128 + lane * 4;
    float s0 = 0.0f, s1 = 0.0f;
#pragma unroll
    for (int q = 0; q < 4; ++q) {
        const float tv = trow[q];
        const int d = lane * 4 + q;
        s0 += tv * Wf2[2 * d + 0];
        s1 += tv * Wf2[2 * d + 1];
    }
#pragma unroll
    for (int off = 16; off > 0; off >>= 1) {
        s0 += __shfl_xor(s0, off, 32);
        s1 += __shfl_xor(s1, off, 32);
    }
    s0 += bf2[0]; s1 += bf2[1];
    const float m = fmaxf(s0, s1);
    const float e0 = __expf(s0 - m), e1 = __expf(s1 - m);
    const float inv = 1.0f / (e0 + e1);
    const float a0 = e0 * inv, a1 = e1 * inv;
    const float4 cb = *(const float4*)(collab  + (size_t)r * 128 + lane * 4);
    const float4 cc = *(const float4*)(content + (size_t)r * 128 + lane * 4);
    float4 o;
    o.x = a0 * (cb.x * (1.0f / 3.0f)) + a1 * cc.x;
    o.y = a0 * (cb.y * (1.0f / 3.0f)) + a1 * cc.y;
    o.z = a0 * (cb.z * (1.0f / 3.0f)) + a1 * cc.z;
    o.w = a0 * (cb.w * (1.0f / 3.0f)) + a1 * cc.w;
    *(float4*)(fused + (size_t)r * 128 + lane * 4) = o;
}

// One wave per prediction pair: score = dot(acc_user[u]/3, fused[b])
__global__ void score_kernel(const float* __restrict__ acc_user, const float* __restrict__ fused,
                             const int* __restrict__ pu, const int* __restrict__ pb,
                             float* __restrict__ out, int P)
{
    const int lane = threadIdx.x & 31;
    const int p = (int)(((size_t)blockIdx.x * blockDim.x + threadIdx.x) >> 5);
    if (p >= P) return;
    const int u = pu[p], b = pb[p];
    const float4 a = *(const float4*)(acc_user + (size_t)u * 128 + lane * 4);
    const float4 f = *(const float4*)(fused    + (size_t)b * 128 + lane * 4);
    float s = a.x * f.x + a.y * f.y + a.z * f.z + a.w * f.w;
#pragma unroll
    for (int off = 16; off > 0; off >>= 1) s += __shfl_xor(s, off, 32);
    if (lane == 0) out[p] = s * (1.0f / 3.0f);
}

// ---------------------------------------------------------------------------
extern "C" void kernel_launch(void* const* d_in, const int* in_sizes, int n_in,
                              void* d_out, int out_size, void* d_ws, size_t ws_size,
                              hipStream_t stream)
{
    const float* user_x = (const float*)d_in[0];
    const float* book_x = (const float*)d_in[1];
    const int*   ub_src = (const int*)d_in[2];
    const int*   ub_dst = (const int*)d_in[3];
    const int*   bb_src = (const int*)d_in[4];
    const int*   bb_dst = (const int*)d_in[5];
    const int*   pred_u = (const int*)d_in[6];
    const int*   pred_b = (const int*)d_in[7];
    const float* Wu  = (const float*)d_in[8];
    const float* bu  = (const float*)d_in[9];
    const float* Wub = (const float*)d_in[10];
    const float* bub = (const float*)d_in[11];
    const float* Wbb = (const float*)d_in[12];
    const float* bbb = (const float*)d_in[13];
    const float* Wl1 = (const float*)d_in[14];
    const float* bl1 = (const float*)d_in[15];
    const float* Wr1 = (const float*)d_in[16];
    const float* Wl2 = (const float*)d_in[17];
    const float* bl2 = (const float*)d_in[18];
    const float* Wr2 = (const float*)d_in[19];
    const float* Wf1 = (const float*)d_in[20];
    const float* bf1 = (const float*)d_in[21];
    const float* Wf2 = (const float*)d_in[22];
    const float* bf2 = (const float*)d_in[23];

    const int DU = 64, H = 128;
    const int U  = in_sizes[0] / DU;
    const int Bn = in_sizes[1] / H;
    const int N  = U + Bn;
    const int E_UB = in_sizes[2];
    const int E_BB = in_sizes[4];
    const int P    = in_sizes[6];

    // ---- workspace carve (floats) ----
    float* ws   = (float*)d_ws;
    float* ACC  = ws;                               // N*128 : x0, then acc
    float* CUR  = ACC  + (size_t)N * 128;           // N*128
    float* NXT  = CUR  + (size_t)N * 128;           // N*128
    float* HB0  = NXT  + (size_t)N * 128;           // B*128
    float* HB1  = HB0  + (size_t)Bn * 128;          // B*128
    float* AGG  = HB1  + (size_t)Bn * 128;          // B*128
    float* TBUF = AGG  + (size_t)Bn * 128;          // B*128
    float* FUS  = TBUF + (size_t)Bn * 128;          // B*128
    float* DEG  = FUS  + (size_t)Bn * 128;          // N
    float* DINV = DEG  + N;                         // N
    float* DEGB = DINV + N;                         // B
    float* IDB  = DEGB + Bn;                        // B

    const size_t rowBytesN = (size_t)N * 128 * sizeof(float);
    const size_t rowBytesB = (size_t)Bn * 128 * sizeof(float);

    auto cdiv = [](long a, long b) { return (int)((a + b - 1) / b); };
    const int TB = 256;

    // ---- degrees ----
    hipMemsetAsync(DEG, 0, (size_t)(2 * N + 2 * Bn) * sizeof(float), stream);
    deg_ub_kernel<<<cdiv(E_UB, TB), TB, 0, stream>>>(ub_src, ub_dst, DEG, E_UB, U);
    count_kernel <<<cdiv(E_BB, TB), TB, 0, stream>>>(bb_dst, DEGB, E_BB);
    dinv_kernel  <<<cdiv(N, TB), TB, 0, stream>>>(DEG, DINV, N);
    invmax1_kernel<<<cdiv(Bn, TB), TB, 0, stream>>>(DEGB, IDB, Bn);

    // ---- input GEMMs: x0 and h0 (fp32 WMMA) ----
    gemm128_wmma<<<cdiv(U / 16, 4), 128, 64 * 128 * 4, stream>>>(
        user_x, 64, Wu, nullptr, 0, nullptr, bu, nullptr, 1.0f, 0, ACC, U);
    gemm128_wmma<<<cdiv(Bn / 16, 4), 128, 128 * 128 * 4, stream>>>(
        book_x, 128, Wub, nullptr, 0, nullptr, bub, nullptr, 1.0f, 0,
        ACC + (size_t)U * 128, Bn);
    gemm128_wmma<<<cdiv(Bn / 16, 4), 128, 128 * 128 * 4, stream>>>(
        book_x, 128, Wbb, nullptr, 0, nullptr, bbb, nullptr, 1.0f, 0, HB0, Bn);

    // ---- LightGCN: 2 hops, acc = x0 + x1 + x2 ----
    hipMemcpyAsync(CUR, ACC, rowBytesN, hipMemcpyDeviceToDevice, stream);
    const int propBlocks = cdiv((long)E_UB * 32, TB);
    const size_t n4 = (size_t)N * 32;

    hipMemsetAsync(NXT, 0, rowBytesN, stream);
    prop_ub_kernel<<<propBlocks, TB, 0, stream>>>(CUR, NXT, ub_src, ub_dst, DINV, E_UB, U);
    vec_add_kernel<<<cdiv((long)n4, TB), TB, 0, stream>>>(ACC, NXT, n4);

    hipMemsetAsync(CUR, 0, rowBytesN, stream);
    prop_ub_kernel<<<propBlocks, TB, 0, stream>>>(NXT, CUR, ub_src, ub_dst, DINV, E_UB, U);
    vec_add_kernel<<<cdiv((long)n4, TB), TB, 0, stream>>>(ACC, CUR, n4);

    // ---- content GNN: 2 mean-agg layers ----
    const int scatBlocks = cdiv((long)E_BB * 32, TB);
    hipMemsetAsync(AGG, 0, rowBytesB, stream);
    scat_bb_kernel<<<scatBlocks, TB, 0, stream>>>(HB0, AGG, bb_src, bb_dst, E_BB);
    gemm128_wmma<<<cdiv(Bn / 16, 4), 128, 128 * 128 * 4, stream>>>(
        AGG, 128, Wl1, HB0, 128, Wr1, bl1, IDB, 1.0f, 1, HB1, Bn);

    hipMemsetAsync(AGG, 0, rowBytesB, stream);
    scat_bb_kernel<<<scatBlocks, TB, 0, stream>>>(HB1, AGG, bb_src, bb_dst, E_BB);
    gemm128_wmma<<<cdiv(Bn / 16, 4), 128, 128 * 128 * 4, stream>>>(
        AGG, 128, Wl2, HB1, 128, Wr2, bl2, IDB, 1.0f, 1, HB0, Bn);   // content -> HB0

    // ---- fusion MLP: T = relu(collab/3 @ Wf1[:128] + content @ Wf1[128:] + bf1) ----
    gemm128_wmma<<<cdiv(Bn / 16, 4), 128, 128 * 128 * 4, stream>>>(
        ACC + (size_t)U * 128, 128, Wf1, HB0, 128, Wf1 + 128 * 128,
        bf1, nullptr, 1.0f / 3.0f, 1, TBUF, Bn);

    attn_fuse_kernel<<<cdiv((long)Bn * 32, TB), TB, 0, stream>>>(
        TBUF, Wf2, bf2, ACC + (size_t)U * 128, HB0, FUS, Bn);

    // ---- scores ----
    score_kernel<<<cdiv((long)P * 32, TB), TB, 0, stream>>>(
        ACC, FUS, pred_u, pred_b, (float*)d_out, P);
}